// LOI_13786845020824
// MI455X (gfx1250) — compile-verified
//
#include <hip/hip_runtime.h>

typedef __attribute__((ext_vector_type(2))) float v2f;
typedef __attribute__((ext_vector_type(8))) float v8f;

#define IMG_H 512
#define IMG_W 512
#define N_C   3
#define N_S   2
#define N_A   3
#define N_BB  2
#define N_BINS 8
#define TILE  16
#define SK    9      // sigma kernel size (9x9, halo 4)
#define AK    7      // alpha kernel size (7x7, halo 3)
#define HALO1 4
#define HALO2 3
#define IN_T  (TILE + 2*HALO2 + 2*HALO1)   // 30: input tile rows/cols
#define ISO_T (TILE + 2*HALO2)             // 22: iso / conv1-output tile
#define ISO_P 24                           // padded to K-chunk multiple of 4

// Fused: clip -> 9x9 sep-Gauss -> soft-bin (exp) -> 7x7 sep-Gauss (via f32 WMMA
// banded matmuls) -> transposed coalesced store. One block = one 16x16 tile of
// one (beta,sigma,channel) combo; wave w owns bin w (8 waves, 8 bins).
__global__ __launch_bounds__(256)
void loi_fused_kernel(const float* __restrict__ im,     // (512,512,3) HWC
                      const float* __restrict__ sigk,   // (2,9,9)
                      const float* __restrict__ alpk,   // (3,7,7)
                      const float* __restrict__ ctrs,   // (8)
                      const float* __restrict__ betas,  // (2)
                      float* __restrict__ out)          // (3,2,2,3,512,512,8)
{
  __shared__ float imT[IN_T][32];                      // clipped input tile
  __shared__ float htmp[IN_T][ISO_P];                  // conv1 horizontal pass
  __shared__ float convT[ISO_T][ISO_P];                // conv1 output
  __shared__ float isoB[N_BINS][ISO_P][ISO_P];         // soft-bin responses (zero padded)
  __shared__ float tmpV[8][TILE][ISO_P];               // per-wave vertical-pass result
  __shared__ __align__(16) float outT[TILE][TILE][N_BINS]; // (h,w,n) == global layout
  __shared__ float sv[SK];                             // 1-D sigma taps
  __shared__ float av[N_A][AK];                        // 1-D alpha taps
  __shared__ float ctr[N_BINS];
  __shared__ float bc[2];                              // 0: 0.5/beta^2, 1: 1/(sqrt(2pi)*beta)

  const int tid  = threadIdx.x;
  const int lane = tid & 31;
  const int wv   = tid >> 5;            // wave id == bin id
  const int th   = blockIdx.x >> 5;
  const int tw   = blockIdx.x & 31;
  const int h0   = th * TILE, w0 = tw * TILE;
  const int bsc  = blockIdx.y;          // 0..11
  const int c    = bsc % 3;
  const int s    = (bsc / 3) & 1;
  const int b    = bsc / 6;

  // ---- parameters -> LDS (1-D taps recovered from normalized 2-D kernels) ----
  if (tid < SK)
    sv[tid] = sigk[s*81 + tid*9 + 4] * rsqrtf(sigk[s*81 + 40]);
  if (tid >= 32 && tid < 32 + N_A*AK) {
    int q = tid - 32;
    av[q/AK][q%AK] = alpk[(q/AK)*49 + (q%AK)*7 + 3] * rsqrtf(alpk[(q/AK)*49 + 24]);
  }
  if (tid >= 64 && tid < 64 + N_BINS)
    ctr[tid - 64] = ctrs[tid - 64];
  if (tid == 96) {
    float bv = betas[b];
    bc[0] = 0.5f / (bv * bv);
    bc[1] = 0.3989422804014327f / bv;   // 1/sqrt(2*pi)/beta
  }

  // ---- load clipped input tile (zero outside image: conv1 zero-pads) ----
  for (int i = tid; i < IN_T*32; i += 256) {
    int r = i >> 5, cc = i & 31;
    float v = 0.f;
    if (cc < IN_T) {
      int gh = h0 - HALO2 - HALO1 + r;
      int gw = w0 - HALO2 - HALO1 + cc;
      if (gh >= 0 && gh < IMG_H && gw >= 0 && gw < IMG_W) {
        float x = im[((size_t)gh*IMG_W + gw)*N_C + c];
        v = fminf(fmaxf(x, 0.f), 1.f);
      }
    }
    imT[r][cc] = v;
  }
  __syncthreads();

  // ---- conv1 horizontal (9 taps) ----
  for (int i = tid; i < IN_T*ISO_T; i += 256) {
    int r = i / ISO_T, cc = i % ISO_T;
    float acc = 0.f;
    #pragma unroll
    for (int t = 0; t < SK; ++t) acc = fmaf(imT[r][cc + t], sv[t], acc);
    htmp[r][cc] = acc;
  }
  __syncthreads();

  // ---- conv1 vertical (9 taps) ----
  for (int i = tid; i < ISO_T*ISO_T; i += 256) {
    int rr = i / ISO_T, cc = i % ISO_T;
    float acc = 0.f;
    #pragma unroll
    for (int t = 0; t < SK; ++t) acc = fmaf(htmp[rr + t][cc], sv[t], acc);
    convT[rr][cc] = acc;
  }
  __syncthreads();

  // ---- soft binning: iso = exp(-0.5 ((conv-ctr)/beta)^2)/(sqrt(2pi) beta) ----
  // zero outside the image (conv2 zero-pads its input) and in the K/N padding
  for (int i = tid; i < N_BINS*ISO_P*ISO_P; i += 256) {
    int n   = i / (ISO_P*ISO_P);
    int rem = i % (ISO_P*ISO_P);
    int r = rem / ISO_P, cc = rem % ISO_P;
    float v = 0.f;
    if (r < ISO_T && cc < ISO_T) {
      int gh = h0 - HALO2 + r, gw = w0 - HALO2 + cc;
      if (gh >= 0 && gh < IMG_H && gw >= 0 && gw < IMG_W) {
        float d = convT[r][cc] - ctr[n];
        v = bc[1] * __expf(-bc[0] * d * d);
      }
    }
    isoB[n][r][cc] = v;
  }
  __syncthreads();

  // ---- conv2 (7x7 separable) as banded f32 WMMA matmuls; wave wv = bin wv ----
  // A 16x4 f32 layout: lane&15 = row M; lanes<16 hold K 0,1; lanes>=16 hold K 2,3.
  // C/D 16x16 f32: VGPR r holds row r (lanes 0-15) / row r+8 (lanes 16-31).
  const int m   = lane & 15;
  const int kkb = (lane < 16) ? 0 : 2;

  for (int a = 0; a < N_A; ++a) {
    // Banded operand Band[k, j] = av[a][k - j] (0 <= k-j < 7, else 0).
    // Identical per-lane values serve as A (vertical pass) and B (horizontal).
    v2f band[6];
    #pragma unroll
    for (int kc = 0; kc < 6; ++kc) {
      int r0 = kc*4 + kkb;
      int d0 = r0 - m, d1 = r0 + 1 - m;
      v2f e;
      e.x = (d0 >= 0 && d0 < AK) ? av[a][d0] : 0.f;
      e.y = (d1 >= 0 && d1 < AK) ? av[a][d1] : 0.f;
      band[kc] = e;
    }

    // Vertical: tmpV[h][w] = sum_t av[t]*iso[h+t][w]  -> D = Band(16x24) x Iso(24x16)
    #pragma unroll
    for (int wc = 0; wc < 2; ++wc) {      // column chunks 0..15 and 8..23
      const int wcb = wc * 8;
      v8f acc = {};
      #pragma unroll
      for (int kc = 0; kc < 6; ++kc) {
        int r0 = kc*4 + kkb;
        v2f bb;
        bb.x = isoB[wv][r0    ][wcb + m];
        bb.y = isoB[wv][r0 + 1][wcb + m];
        acc = __builtin_amdgcn_wmma_f32_16x16x4_f32(
                  false, band[kc], false, bb, (short)0, acc, false, false);
      }
      #pragma unroll
      for (int rr = 0; rr < 8; ++rr) {
        int row = (lane < 16) ? rr : rr + 8;
        tmpV[wv][row][wcb + m] = acc[rr];
      }
    }

    // Horizontal: out[h][w] = sum_t av[t]*tmp[h][w+t] -> D = Tmp(16x24) x Band(24x16)
    {
      v8f acc = {};
      #pragma unroll
      for (int kc = 0; kc < 6; ++kc) {
        int r0 = kc*4 + kkb;                       // even -> 8B aligned float2
        v2f aa = *(const v2f*)&tmpV[wv][m][r0];
        acc = __builtin_amdgcn_wmma_f32_16x16x4_f32(
                  false, aa, false, band[kc], (short)0, acc, false, false);
      }
      #pragma unroll
      for (int rr = 0; rr < 8; ++rr) {
        int row = (lane < 16) ? rr : rr + 8;
        outT[row][m][wv] = acc[rr];
      }
    }
    __syncthreads();

    // ---- coalesced store: each h-row is 128 contiguous floats (w0..w0+15, n0..7) ----
    {
      size_t base = ((((size_t)a*N_BB + b)*N_S + s)*N_C + c) * ((size_t)IMG_H*IMG_W*N_BINS);
      const float* src = &outT[0][0][0];
      #pragma unroll
      for (int v = 0; v < 2; ++v) {
        int flat = (tid*2 + v) * 4;       // 2048 floats / block / alpha
        int hh   = flat >> 7;
        int rem  = flat & 127;
        size_t gidx = base + (size_t)(h0 + hh)*(IMG_W*N_BINS) + (size_t)w0*N_BINS + rem;
        *(float4*)(out + gidx) = *(const float4*)(src + flat);
      }
    }
    __syncthreads();
  }
}

extern "C" void kernel_launch(void* const* d_in, const int* in_sizes, int n_in,
                              void* d_out, int out_size, void* d_ws, size_t ws_size,
                              hipStream_t stream) {
  (void)in_sizes; (void)n_in; (void)d_ws; (void)ws_size; (void)out_size;
  const float* im   = (const float*)d_in[0];  // im_th (512,512,3)
  const float* sigk = (const float*)d_in[1];  // sigma_kernels (2,9,9)
  const float* alpk = (const float*)d_in[2];  // alpha_kernels (3,7,7)
  const float* ctrs = (const float*)d_in[3];  // bin_centers (8)
  const float* bet  = (const float*)d_in[4];  // betas (2)
  float* out = (float*)d_out;                 // (3,2,2,3,512,512,8) fp32

  dim3 grid((IMG_H/TILE) * (IMG_W/TILE), N_BB * N_S * N_C);  // 1024 x 12
  loi_fused_kernel<<<grid, 256, 0, stream>>>(im, sigk, alpk, ctrs, bet, out);
}